// SAGE_90134183674598
// MI455X (gfx1250) — compile-verified
//
#include <hip/hip_runtime.h>
#include <hip/hip_bf16.h>

typedef __attribute__((ext_vector_type(2))) float v2f;
typedef __attribute__((ext_vector_type(8))) float v8f;
typedef __attribute__((ext_vector_type(4))) int   v4i;

#define N_NODES 100000
#define N_EDGES 800000
#define IN_SIZE 256
#define HID 128
#define OUT_SIZE 64
#define MAXK_K 32
#define LDS_PAD 4   // row pad (floats) so A-fragment ds_loads spread across banks

#if __has_builtin(__builtin_amdgcn_global_load_async_to_lds_b128)
#define HAVE_ASYNC_LDS 1
#else
#define HAVE_ASYNC_LDS 0
#endif

__device__ __forceinline__ void wait_async_and_barrier() {
#if HAVE_ASYNC_LDS
#if __has_builtin(__builtin_amdgcn_s_wait_asynccnt)
    __builtin_amdgcn_s_wait_asynccnt(0);
#else
    asm volatile("s_wait_asynccnt 0x0" ::: "memory");
#endif
#endif
    __syncthreads();
}

// Copy a 16-row x K-col f32 tile (contiguous in global: 16 consecutive rows)
// into LDS with row stride K+LDS_PAD, using async global->LDS b128 transfers.
__device__ __forceinline__ void stage_tile(const float* __restrict__ g, float* lds,
                                           int K, int tid, int nthr) {
    const int cpr = K >> 2;            // 16B chunks per row
    const int total = 16 * cpr;
    for (int c = tid; c < total; c += nthr) {
        const int row = c / cpr;
        const int col = (c - row * cpr) << 2;       // float index within row
        const float* gp = g + (long)row * K + col;
        float* lp = lds + row * (K + LDS_PAD) + col;
#if HAVE_ASYNC_LDS
        __builtin_amdgcn_global_load_async_to_lds_b128(
            (v4i*)gp, (v4i*)lp, /*imm offset*/ 0, /*cpol*/ 0);
#else
        *(float4*)lp = *(const float4*)gp;
#endif
    }
}

// ---------------------------------------------------------------------------
// fp32 WMMA GEMM:  C[M,N] = A[M,K] @ B[K,N] (+ A2[M,K] @ B2[K,N]) (+ bias)
// One wave computes one 16x16 tile via V_WMMA_F32_16X16X4_F32, A staged in LDS.
// Two independent accumulator chains per wave (K halves, or GEMM1/GEMM2).
// block = (32, N/16), grid.x = M/16. Requires K2 == K when A2 != null.
//
// Fragment layouts per CDNA5 ISA 7.12.2 (32-bit, wave32):
//   A 16x4 : lanes 0-15 -> M=lane, v0=K0,v1=K1 ; lanes 16-31 -> v0=K2,v1=K3
//   B 4x16 : v0 -> K=0 (lanes 0-15) / K=2 (lanes 16-31), v1 -> K=1 / K=3
//   C/D    : vr -> M=r (lanes 0-15) / M=r+8 (lanes 16-31), N=lane%16
// ---------------------------------------------------------------------------
__global__ void sage_wmma_gemm(const float* __restrict__ A, const float* __restrict__ B,
                               const float* __restrict__ A2, const float* __restrict__ B2,
                               const float* __restrict__ bias, float* __restrict__ C,
                               int N, int K, int K2) {
    __shared__ float ldsA [16 * (IN_SIZE + LDS_PAD)];   // 16.6 KB (worst case K=256)
    __shared__ float ldsA2[16 * (HID    + LDS_PAD)];    //  8.4 KB

    const int lane = threadIdx.x;          // 0..31
    const int half = lane >> 4;            // 0 or 1
    const int lm   = lane & 15;
    const long row0 = (long)blockIdx.x * 16;
    const int  n    = threadIdx.y * 16 + lm;
    const int  tid  = threadIdx.y * 32 + lane;
    const int  nthr = blockDim.y * 32;

    stage_tile(A + row0 * K, ldsA, K, tid, nthr);
    if (A2 != nullptr) stage_tile(A2 + row0 * K2, ldsA2, K2, tid, nthr);
    wait_async_and_barrier();

    v8f acc0 = {}, acc1 = {};
    if (A2 != nullptr) {                    // fused dual GEMM, K2 == K
        const float* __restrict__ ar  = ldsA  + lm * (K  + LDS_PAD) + 2 * half;
        const float* __restrict__ ar2 = ldsA2 + lm * (K2 + LDS_PAD) + 2 * half;
        for (int k = 0; k < K; k += 4) {
            v2f a, b;
            a.x = ar[k];  a.y = ar[k + 1];
            b.x = B [(long)(k + 2 * half) * N + n];
            b.y = B [(long)(k + 2 * half + 1) * N + n];
            acc0 = __builtin_amdgcn_wmma_f32_16x16x4_f32(false, a, false, b,
                                                         (short)0, acc0, false, false);
            a.x = ar2[k]; a.y = ar2[k + 1];
            b.x = B2[(long)(k + 2 * half) * N + n];
            b.y = B2[(long)(k + 2 * half + 1) * N + n];
            acc1 = __builtin_amdgcn_wmma_f32_16x16x4_f32(false, a, false, b,
                                                         (short)0, acc1, false, false);
        }
    } else {                                // single GEMM: split K into 2 chains
        const float* __restrict__ ar = ldsA + lm * (K + LDS_PAD) + 2 * half;
        const int Kh = K >> 1;
        for (int k = 0; k < Kh; k += 4) {
            v2f a, b;
            a.x = ar[k];  a.y = ar[k + 1];
            b.x = B[(long)(k + 2 * half) * N + n];
            b.y = B[(long)(k + 2 * half + 1) * N + n];
            acc0 = __builtin_amdgcn_wmma_f32_16x16x4_f32(false, a, false, b,
                                                         (short)0, acc0, false, false);
            a.x = ar[k + Kh]; a.y = ar[k + Kh + 1];
            b.x = B[(long)(k + Kh + 2 * half) * N + n];
            b.y = B[(long)(k + Kh + 2 * half + 1) * N + n];
            acc1 = __builtin_amdgcn_wmma_f32_16x16x4_f32(false, a, false, b,
                                                         (short)0, acc1, false, false);
        }
    }

    const float bv = (bias != nullptr) ? bias[n] : 0.0f;
#pragma unroll
    for (int r = 0; r < 8; ++r) {
        C[(row0 + r + 8 * half) * (long)N + n] = acc0[r] + acc1[r] + bv;
    }
}

// ---------------------------------------------------------------------------
// MaxK nonlinearity: keep top-32 of each 128-wide row, zero the rest.
// Tie-break identical to jax.lax.top_k: value desc, then index asc.
// One wave per row; 8 rows per 256-thread block via LDS.
// ---------------------------------------------------------------------------
__global__ void sage_maxk(const float* __restrict__ h, float* __restrict__ o) {
    __shared__ float row[8][HID];
    const int r = blockIdx.x * 8 + threadIdx.y;
    const float* __restrict__ hr = h + (long)r * HID;

    for (int j = threadIdx.x; j < HID; j += 32)
        row[threadIdx.y][j] = hr[j];
    __syncthreads();

    for (int j = threadIdx.x; j < HID; j += 32) {
        const float v = row[threadIdx.y][j];
        int rank = 0;
#pragma unroll 8
        for (int t = 0; t < HID; ++t) {
            const float u = row[threadIdx.y][t];
            rank += (u > v) || (u == v && t < j);
        }
        o[(long)r * HID + j] = (rank < MAXK_K) ? v : 0.0f;
    }
}

// ---------------------------------------------------------------------------
// Edge scatter: agg[src] += h[dst] (128 f32 atomics / edge), deg[src] += 1.
// One wave per edge; lane j gathers float4 (16B) of the message row.
// ---------------------------------------------------------------------------
__global__ void sage_scatter(const float* __restrict__ h,
                             const int* __restrict__ src, const int* __restrict__ dst,
                             float* __restrict__ agg, float* __restrict__ deg, int E) {
    const int e = blockIdx.x * 8 + threadIdx.y;
    if (e >= E) return;
    const int s = src[e];
    const int d = dst[e];
    const float4 v = ((const float4*)(h + (long)d * HID))[threadIdx.x];
    float* ap = agg + (long)s * HID + threadIdx.x * 4;
    atomicAdd(ap + 0, v.x);
    atomicAdd(ap + 1, v.y);
    atomicAdd(ap + 2, v.z);
    atomicAdd(ap + 3, v.w);
    if (threadIdx.x == 0) atomicAdd(deg + s, 1.0f);
}

// agg[i][f] /= (deg[i] + 1e-6)
__global__ void sage_normalize(float* __restrict__ agg, const float* __restrict__ deg) {
    const long i = (long)blockIdx.x * blockDim.x + threadIdx.x;
    if (i < (long)N_NODES * HID) {
        agg[i] = agg[i] / (deg[i >> 7] + 1e-6f);
    }
}

extern "C" void kernel_launch(void* const* d_in, const int* in_sizes, int n_in,
                              void* d_out, int out_size, void* d_ws, size_t ws_size,
                              hipStream_t stream) {
    const float* x        = (const float*)d_in[0];
    const float* W_in     = (const float*)d_in[1];
    const float* b_in     = (const float*)d_in[2];
    const float* W_self0  = (const float*)d_in[3];
    const float* W_neigh0 = (const float*)d_in[4];
    const float* W_self1  = (const float*)d_in[5];
    const float* W_neigh1 = (const float*)d_in[6];
    const float* W_out    = (const float*)d_in[7];
    const float* b_out    = (const float*)d_in[8];
    const int*   e_src    = (const int*)d_in[9];
    const int*   e_dst    = (const int*)d_in[10];
    float*       out      = (float*)d_out;

    const long NH = (long)N_NODES * HID;           // 12.8M floats
    float* h0  = (float*)d_ws;                     // [N, HID]
    float* hs  = h0  + NH;                         // sparse (maxk) activations
    float* agg = hs  + NH;                         // aggregation accumulator
    float* h1  = agg + NH;                         // layer output
    float* deg = h1  + NH;                         // [N]

    const dim3 gemm_grid(N_NODES / 16);
    const dim3 gemm_blk_hid(32, HID / 16);         // 256 threads, 8 waves
    const dim3 gemm_blk_out(32, OUT_SIZE / 16);    // 128 threads, 4 waves
    const dim3 maxk_grid(N_NODES / 8), maxk_blk(32, 8);
    const dim3 sc_grid((N_EDGES + 7) / 8), sc_blk(32, 8);
    const dim3 nrm_grid((NH + 255) / 256), nrm_blk(256);

    // h0 = x @ W_in + b_in
    sage_wmma_gemm<<<gemm_grid, gemm_blk_hid, 0, stream>>>(
        x, W_in, nullptr, nullptr, b_in, h0, HID, IN_SIZE, 0);

    const float* Ws[2] = {W_self0, W_self1};
    const float* Wn[2] = {W_neigh0, W_neigh1};
    float* hin  = h0;
    float* hout = h1;
    for (int l = 0; l < 2; ++l) {
        // hs = maxk(hin, 32)
        sage_maxk<<<maxk_grid, maxk_blk, 0, stream>>>(hin, hs);
        // agg = segment_sum(hs[dst] -> src); deg = segment_sum(1)
        (void)hipMemsetAsync(agg, 0, NH * sizeof(float), stream);
        (void)hipMemsetAsync(deg, 0, (size_t)N_NODES * sizeof(float), stream);
        sage_scatter<<<sc_grid, sc_blk, 0, stream>>>(hs, e_src, e_dst, agg, deg, N_EDGES);
        sage_normalize<<<nrm_grid, nrm_blk, 0, stream>>>(agg, deg);
        // hout = hs @ W_self + agg @ W_neigh   (fused dual-GEMM)
        sage_wmma_gemm<<<gemm_grid, gemm_blk_hid, 0, stream>>>(
            hs, Ws[l], agg, Wn[l], nullptr, hout, HID, HID, HID);
        float* t = hin; hin = hout; hout = t;      // ping-pong
    }

    // out = hin @ W_out + b_out
    sage_wmma_gemm<<<gemm_grid, gemm_blk_out, 0, stream>>>(
        hin, W_out, nullptr, nullptr, b_out, out, OUT_SIZE, HID, 0);
}